// GraphSAGEModel_84670985273715
// MI455X (gfx1250) — compile-verified
//
#include <hip/hip_runtime.h>

typedef __attribute__((ext_vector_type(2))) float v2f;
typedef __attribute__((ext_vector_type(8))) float v8f;

#define N_NODES 100000
#define N_EDGES 600000
#define FEAT    128          // feature width of every GEMM input (in & hidden)

// ---------------------------------------------------------------- zero fill
__global__ void zero_kernel(float* __restrict__ p, long n4) {
  long i = ((long)blockIdx.x * blockDim.x + threadIdx.x);
  if (i < n4) {
    float4 z = make_float4(0.f, 0.f, 0.f, 0.f);
    *(float4*)(p + i * 4) = z;
  }
}

// ---------------------------------------------------------------- degree
__global__ void deg_kernel(const int* __restrict__ dst, float* __restrict__ deg) {
  int e = blockIdx.x * blockDim.x + threadIdx.x;
  if (e < N_EDGES) atomicAdd(&deg[dst[e]], 1.0f);
}

// ------------------------------------------------- edge gather + scatter-add
// 32 lanes per edge, float4 per lane: coalesced 512B gather per edge,
// 4 f32 atomic adds per lane into msg (L2-resident, L2 atomic units do RMW).
__global__ void aggregate_kernel(const float* __restrict__ h,
                                 const int* __restrict__ src,
                                 const int* __restrict__ dst,
                                 float* __restrict__ msg) {
  long gid = (long)blockIdx.x * blockDim.x + threadIdx.x;
  int e = (int)(gid >> 5);
  if (e >= N_EDGES) return;
  int off = (int)(gid & 31) * 4;
  const float4 v = *(const float4*)(h + (long)src[e] * FEAT + off);
  float* m = msg + (long)dst[e] * FEAT + off;
  atomicAdd(m + 0, v.x);
  atomicAdd(m + 1, v.y);
  atomicAdd(m + 2, v.z);
  atomicAdd(m + 3, v.w);
}

// one K=4 step: A fragment 'a' vs 4 adjacent 16-col tiles; each B fragment is a
// single aligned ds_load_b64 thanks to the pair-interleaved LDS weight layout.
#define WMMA4(a, wl, c0v, c1v, c2v, c3v)                                                            \
  do {                                                                                              \
    c0v = __builtin_amdgcn_wmma_f32_16x16x4_f32(false, a, false, *(const v2f*)((wl) + 0),  (short)0, c0v, false, false); \
    c1v = __builtin_amdgcn_wmma_f32_16x16x4_f32(false, a, false, *(const v2f*)((wl) + 32), (short)0, c1v, false, false); \
    c2v = __builtin_amdgcn_wmma_f32_16x16x4_f32(false, a, false, *(const v2f*)((wl) + 64), (short)0, c2v, false, false); \
    c3v = __builtin_amdgcn_wmma_f32_16x16x4_f32(false, a, false, *(const v2f*)((wl) + 96), (short)0, c3v, false, false); \
  } while (0)

// ------------------------------------------------------------- fused GEMM
// out[N_NODES x Ncols] = act( [h | msg/deg] @ [Wself ; Wneigh] + bself+bneigh )
// KT = 256 (with neighbor path) or 128 (final linear). Block = 256 thr = 8 waves,
// each wave owns a 16-row strip; weight chunk staged in 64KB LDS with K-pair
// interleave: element (k,c) -> wlds[(k>>1)*128 + c*2 + (k&1)], so the WMMA B
// fragment (W[ka][n], W[ka+1][n]) is one contiguous 8B ds_load_b64.
__global__ void sage_gemm_kernel(const float* __restrict__ h,
                                 const float* __restrict__ msg,     // null if KT==128
                                 const float* __restrict__ deg,
                                 const float* __restrict__ Wself,
                                 const float* __restrict__ Wneigh,  // unused if KT==128
                                 const float* __restrict__ bself,
                                 const float* __restrict__ bneigh,  // may be null
                                 float* __restrict__ out,
                                 int Ncols, int KT, int do_relu) {
  __shared__ float wlds[256 * 64];                       // 64 KB chunk
  const int K1   = FEAT;
  const int tid  = threadIdx.x;
  const int wave = tid >> 5;
  const int lane = tid & 31;
  const int m    = lane & 15;                            // row within A tile / col within B tile
  const int half = lane >> 4;                            // K sub-offset select
  const int row0 = (blockIdx.x * 8 + wave) * 16;
  const int row  = row0 + m;
  const bool active = (row0 < N_NODES);                  // wave-uniform -> EXEC all-1s in WMMA

  // per-lane A base pointers (row fixed per lane; K sub-offset folded in)
  const float* ap = h + (active ? (long)row * FEAT : 0) + 2 * half;
  const float* mp = msg ? (msg + (active ? (long)row * FEAT : 0) + 2 * half) : nullptr;
  float rdeg = 0.f;
  if (msg && active) rdeg = 1.0f / fmaxf(deg[row], 1.0f);

  if (active) {                                          // warm L0/L2 for the A strip
    __builtin_prefetch(ap, 0, 0);
    if (mp) __builtin_prefetch(mp, 0, 0);
  }

  for (int c0 = 0; c0 < Ncols; c0 += 64) {
    __syncthreads();
    for (int i = tid; i < KT * 64; i += 256) {           // stage weight chunk (pair-interleaved)
      int k = i >> 6, c = i & 63;
      float w = (k < K1) ? Wself[(long)k * Ncols + c0 + c]
                         : Wneigh[(long)(k - K1) * Ncols + c0 + c];
      wlds[(k >> 1) * 128 + c * 2 + (k & 1)] = w;
    }
    __syncthreads();
    if (!active) continue;

    v8f acc0, acc1, acc2, acc3;                          // 4 col-tiles of 16
    {
      float b0 = bself[c0 +  0 + m] + (bneigh ? bneigh[c0 +  0 + m] : 0.f);
      float b1 = bself[c0 + 16 + m] + (bneigh ? bneigh[c0 + 16 + m] : 0.f);
      float b2 = bself[c0 + 32 + m] + (bneigh ? bneigh[c0 + 32 + m] : 0.f);
      float b3 = bself[c0 + 48 + m] + (bneigh ? bneigh[c0 + 48 + m] : 0.f);
      for (int j = 0; j < 8; ++j) { acc0[j] = b0; acc1[j] = b1; acc2[j] = b2; acc3[j] = b3; }
    }

    // ---- self half: K in [0,128), software-pipelined A fragments ----
    // A fragment (ISA 16x4 f32 layout): lanes 0-15 -> K=k0,k0+1 ; lanes 16-31 -> K=k0+2,k0+3
    {
      const float* wb = wlds + half * 128 + 2 * m;       // pair row (ka>>1), col n*2
      v2f a = *(const v2f*)(ap);
#pragma unroll 4
      for (int k0 = 0; k0 < K1 - 4; k0 += 4) {
        v2f an = *(const v2f*)(ap + k0 + 4);             // prefetch next A fragment
        WMMA4(a, wb + (k0 >> 1) * 128, acc0, acc1, acc2, acc3);
        a = an;
      }
      WMMA4(a, wb + ((K1 - 4) >> 1) * 128, acc0, acc1, acc2, acc3);
    }
    // ---- neighbor half: K in [128,256), mean-division folded into A ----
    if (KT > K1) {
      const float* wb = wlds + (K1 / 2 + half) * 128 + 2 * m;
      v2f t0 = *(const v2f*)(mp);
      v2f a; a[0] = t0[0] * rdeg; a[1] = t0[1] * rdeg;
#pragma unroll 4
      for (int k0 = 0; k0 < K1 - 4; k0 += 4) {
        v2f tn = *(const v2f*)(mp + k0 + 4);             // prefetch next A fragment
        WMMA4(a, wb + (k0 >> 1) * 128, acc0, acc1, acc2, acc3);
        a[0] = tn[0] * rdeg; a[1] = tn[1] * rdeg;
      }
      WMMA4(a, wb + ((K1 - 4) >> 1) * 128, acc0, acc1, acc2, acc3);
    }

    // C/D layout: VGPR j -> row M=j (lanes 0-15) / M=j+8 (lanes 16-31), col = lane%16
#pragma unroll
    for (int j = 0; j < 8; ++j) {
      long r = row0 + j + 8 * half;
      float* o = out + r * Ncols + c0 + m;
      float x0 = acc0[j], x1 = acc1[j], x2 = acc2[j], x3 = acc3[j];
      if (do_relu) {
        x0 = fmaxf(x0, 0.f); x1 = fmaxf(x1, 0.f);
        x2 = fmaxf(x2, 0.f); x3 = fmaxf(x3, 0.f);
      }
      o[0] = x0; o[16] = x1; o[32] = x2; o[48] = x3;
    }
  }
}

// ---------------------------------------------------------------- launcher
extern "C" void kernel_launch(void* const* d_in, const int* in_sizes, int n_in,
                              void* d_out, int out_size, void* d_ws, size_t ws_size,
                              hipStream_t stream) {
  const float* feats = (const float*)d_in[0];
  const int*   src   = (const int*)d_in[1];
  const int*   dst   = (const int*)d_in[2];
  const float* Ws[3] = {(const float*)d_in[3], (const float*)d_in[7],  (const float*)d_in[11]};
  const float* bs[3] = {(const float*)d_in[4], (const float*)d_in[8],  (const float*)d_in[12]};
  const float* Wn[3] = {(const float*)d_in[5], (const float*)d_in[9],  (const float*)d_in[13]};
  const float* bn[3] = {(const float*)d_in[6], (const float*)d_in[10], (const float*)d_in[14]};
  const float* Wo    = (const float*)d_in[15];
  const float* bo    = (const float*)d_in[16];

  // workspace layout (floats): deg | msg | h1 | h2   (~154 MB total)
  float* ws  = (float*)d_ws;
  float* deg = ws;                                  // 100000, padded to 100352
  float* msg = ws + 100352;                         // 100000*128
  float* h1  = msg + (long)N_NODES * FEAT;
  float* h2  = h1  + (long)N_NODES * FEAT;

  const long msgN   = (long)N_NODES * FEAT;         // 12,800,000 (mult of 4)
  const int  AGG_G  = (int)(((long)N_EDGES * 32 + 255) / 256);
  const int  GEMM_G = (N_NODES / 16 + 7) / 8;       // 6250 strips / 8 waves = 782

  // degree (shared across layers)
  zero_kernel<<<(N_NODES / 4 + 255) / 256, 256, 0, stream>>>(deg, N_NODES / 4);
  deg_kernel<<<(N_EDGES + 255) / 256, 256, 0, stream>>>(dst, deg);

  // layer 0: features -> h1 (relu)
  zero_kernel<<<(int)((msgN / 4 + 255) / 256), 256, 0, stream>>>(msg, msgN / 4);
  aggregate_kernel<<<AGG_G, 256, 0, stream>>>(feats, src, dst, msg);
  sage_gemm_kernel<<<GEMM_G, 256, 0, stream>>>(feats, msg, deg, Ws[0], Wn[0], bs[0], bn[0],
                                               h1, 128, 256, 1);
  // layer 1: h1 -> h2 (relu)
  zero_kernel<<<(int)((msgN / 4 + 255) / 256), 256, 0, stream>>>(msg, msgN / 4);
  aggregate_kernel<<<AGG_G, 256, 0, stream>>>(h1, src, dst, msg);
  sage_gemm_kernel<<<GEMM_G, 256, 0, stream>>>(h1, msg, deg, Ws[1], Wn[1], bs[1], bn[1],
                                               h2, 128, 256, 1);
  // layer 2: h2 -> h1 (no relu)
  zero_kernel<<<(int)((msgN / 4 + 255) / 256), 256, 0, stream>>>(msg, msgN / 4);
  aggregate_kernel<<<AGG_G, 256, 0, stream>>>(h2, src, dst, msg);
  sage_gemm_kernel<<<GEMM_G, 256, 0, stream>>>(h2, msg, deg, Ws[2], Wn[2], bs[2], bn[2],
                                               h1, 128, 256, 0);
  // final linear: h1 @ W_out + b_out -> d_out  (K=128, N=64, no neighbor path)
  sage_gemm_kernel<<<GEMM_G, 256, 0, stream>>>(h1, nullptr, deg, Wo, Wo, bo, nullptr,
                                               (float*)d_out, 64, 128, 0);
}